// Complex_Attention_77927886618995
// MI455X (gfx1250) — compile-verified
//
#include <hip/hip_runtime.h>
#include <hip/hip_bf16.h>

typedef __attribute__((ext_vector_type(2))) float v2f;
typedef __attribute__((ext_vector_type(8))) float v8f;

#define HEAD 16
#define FEAT 32
#define OUTF 32
#define N_E  2048
#define N_UP 1024
#define K1   192   // g*6 + z*2 + b  ==  f*6 + z*2 + b (contiguous per row)

// ---------------------------------------------------------------------------
// Kernel 1: Qp[blk][h][n][k1] = sum_f complex-ish(q, M) per eps table.
//   b0 = sum_f q0*M0 - q1*M1 ;  b1 = sum_f q1*M0 + q0*M1
// blk: 0=uu(q=u) 1=ud(q=u) 2=du(q=d) 3=dd(q=d)
// ---------------------------------------------------------------------------
__global__ void k_qprime(const float* __restrict__ x,
                         const float* __restrict__ Muu, const float* __restrict__ Mdd,
                         const float* __restrict__ Mud, const float* __restrict__ Mdu,
                         float* __restrict__ ws) {
    int tid = blockIdx.x * 256 + threadIdx.x;           // 4*16*1024*96 threads
    int gz  = tid % 96;
    int rem = tid / 96;
    int n   = rem % 1024; rem /= 1024;
    int h   = rem % 16;
    int blk = rem / 16;
    if (blk >= 4) return;
    int g = gz / 3, z = gz % 3;

    const float* q = x + ((blk >> 1) ? (size_t)N_UP * K1 : (size_t)0) + (size_t)n * K1;
    const float* M = (blk == 0) ? Muu : (blk == 1) ? Mud : (blk == 2) ? Mdu : Mdd;
    // M[h][f][g][z][c]: base at f=0, stride per f = 32*3*2 = 192
    const float* Mp = M + (((size_t)h * 32 * 32 + (size_t)g) * 3 + z) * 2;

    float b0 = 0.f, b1 = 0.f;
    #pragma unroll 8
    for (int f = 0; f < FEAT; ++f) {
        float q0 = q[f * 6 + z * 2];
        float q1 = q[f * 6 + z * 2 + 1];
        float M0 = Mp[(size_t)f * 192];
        float M1 = Mp[(size_t)f * 192 + 1];
        b0 = fmaf(q0, M0, fmaf(-q1, M1, b0));
        b1 = fmaf(q1, M0, fmaf( q0, M1, b1));
    }
    float* out = ws + (((size_t)(blk * 16 + h) * 1024 + n) * K1) + g * 6 + z * 2;
    out[0] = b0;
    out[1] = b1;
}

// ---------------------------------------------------------------------------
// Kernel 2: per (block, head, 16-row n-tile):
//   S = Qp(16xK1) x Khat(K1 x 1024)   [WMMA f32 16x16x4, K=192]
//   softmax rows of S
//   W = att(16x1024) x V(1024xK1)     [WMMA f32 16x16x4, K=1024]
//   W overwrites this tile's own Qp rows (already consumed into registers).
// One wave32 per workgroup.
// ---------------------------------------------------------------------------
__global__ __launch_bounds__(32) void k_attn(const float* __restrict__ x,
                                             float* __restrict__ ws) {
    const int nt  = blockIdx.x;      // 0..63
    const int h   = blockIdx.y;      // 0..15
    const int blk = blockIdx.z;      // 0..3
    const int lane = threadIdx.x & 31;
    const int row  = lane & 15;      // A rows / B cols / C cols
    const int hi   = lane >> 4;      // lane half selects K sub-pair / C row half

    // k-side (== v-side): u for blk 0,2 ; d for blk 1,3
    const float* kbase = x + ((blk & 1) ? (size_t)N_UP * K1 : (size_t)0);
    float* QW = ws + (size_t)(blk * 16 + h) * 1024 * K1;   // Qp in, W out (in place)

    __shared__ float S[16][1024];                          // 64 KB

    // ---- preload A fragments: 48 k-steps, v2f each ----
    const float* qrow = QW + (size_t)(nt * 16 + row) * K1 + hi * 2;
    v2f afrag[48];
    #pragma unroll
    for (int k = 0; k < 48; ++k)
        afrag[k] = *(const v2f*)(qrow + 4 * k);

    // ---- scores GEMM: 64 m-tiles ----
    for (int mt = 0; mt < 64; ++mt) {
        v8f acc = {};
        const float* kb = kbase + (size_t)(mt * 16 + row) * K1 + hi * 2;
        #pragma unroll
        for (int k = 0; k < 48; ++k) {
            v2f bf = *(const v2f*)(kb + 4 * k);
            acc = __builtin_amdgcn_wmma_f32_16x16x4_f32(
                      false, afrag[k], false, bf, (short)0, acc, false, false);
        }
        #pragma unroll
        for (int r = 0; r < 8; ++r)
            S[r + 8 * hi][mt * 16 + row] = acc[r];
    }
    __syncthreads();

    // ---- softmax: lanes (r, r+16) handle halves of row r ----
    {
        const int base = hi * 512;
        float mx = -3.0e38f;
        for (int j = 0; j < 512; ++j) mx = fmaxf(mx, S[row][base + j]);
        mx = fmaxf(mx, __shfl_xor(mx, 16, 32));
        float sum = 0.f;
        for (int j = 0; j < 512; ++j) {
            float e = __expf(S[row][base + j] - mx);
            S[row][base + j] = e;
            sum += e;
        }
        sum += __shfl_xor(sum, 16, 32);
        float inv = 1.0f / sum;
        for (int j = 0; j < 512; ++j) S[row][base + j] *= inv;
    }
    __syncthreads();

    // ---- AV GEMM: W(16 x 192) = att(16 x 1024) x V(1024 x 192) ----
    v8f wacc[12];
    #pragma unroll
    for (int t = 0; t < 12; ++t) wacc[t] = (v8f){};

    for (int ms = 0; ms < 256; ++ms) {
        const int m0 = 4 * ms;
        v2f af = *(const v2f*)(&S[row][m0 + 2 * hi]);          // ds_load_b64
        const float* v0 = kbase + (size_t)(m0 + 2 * hi) * K1;
        #pragma unroll
        for (int t = 0; t < 12; ++t) {
            v2f bf;
            bf.x = v0[16 * t + row];
            bf.y = v0[K1 + 16 * t + row];
            wacc[t] = __builtin_amdgcn_wmma_f32_16x16x4_f32(
                          false, af, false, bf, (short)0, wacc[t], false, false);
        }
    }

    // ---- store W over this tile's Qp rows ----
    #pragma unroll
    for (int t = 0; t < 12; ++t) {
        #pragma unroll
        for (int r = 0; r < 8; ++r)
            QW[(size_t)(nt * 16 + r + 8 * hi) * K1 + 16 * t + row] = wacc[t][r];
    }
}

// ---------------------------------------------------------------------------
// Kernel 3: y[n,g,z,c] = sum over {2 blocks} sum_h sum_f  A ⊛ W  per ksi:
//   c0 += A0*W0 - A1*W1 ;  c1 += A0*W1 + A1*W0
// y rows <1024: blocks (0:uu/A_uu, 1:ud/A_ud); rows >=1024: (3:dd/A_dd, 2:du/A_du)
// ---------------------------------------------------------------------------
__global__ void k_combine(const float* __restrict__ ws,
                          const float* __restrict__ Auu, const float* __restrict__ Add,
                          const float* __restrict__ Aud, const float* __restrict__ Adu,
                          float* __restrict__ y) {
    int tid = blockIdx.x * 256 + threadIdx.x;      // 2048*3*32 threads
    int g   = tid % 32;
    int rem = tid / 32;
    int z   = rem % 3;
    int n_out = rem / 3;
    if (n_out >= N_E) return;

    const bool up = (n_out < N_UP);
    const int  n  = up ? n_out : (n_out - N_UP);
    const int  blkA = up ? 0 : 3;
    const int  blkB = up ? 1 : 2;
    const float* A_first  = up ? Auu : Add;
    const float* A_second = up ? Aud : Adu;

    float acc0 = 0.f, acc1 = 0.f;
    for (int p = 0; p < 2; ++p) {
        const int blk = p ? blkB : blkA;
        const float* Ab = p ? A_second : A_first;
        const float* Wb = ws + ((size_t)blk * 16 * 1024 + n) * K1;
        for (int h = 0; h < HEAD; ++h) {
            const float* W  = Wb + (size_t)h * 1024 * K1;
            // A[h][g][f][z][a]: base f=0, stride per f = 6
            const float* Ar = Ab + (((size_t)(h * 32 + g) * 32) * 3 + z) * 2;
            #pragma unroll 8
            for (int f = 0; f < FEAT; ++f) {
                float W0 = W[f * 6 + z * 2];
                float W1 = W[f * 6 + z * 2 + 1];
                float A0 = Ar[f * 6];
                float A1 = Ar[f * 6 + 1];
                acc0 = fmaf(A0, W0, fmaf(-A1, W1, acc0));
                acc1 = fmaf(A0, W1, fmaf( A1, W0, acc1));
            }
        }
    }
    float* o = y + (size_t)n_out * K1 + g * 6 + z * 2;
    o[0] = acc0;
    o[1] = acc1;
}

// ---------------------------------------------------------------------------
extern "C" void kernel_launch(void* const* d_in, const int* in_sizes, int n_in,
                              void* d_out, int out_size, void* d_ws, size_t ws_size,
                              hipStream_t stream) {
    const float* x   = (const float*)d_in[0];
    const float* Muu = (const float*)d_in[1];
    const float* Mdd = (const float*)d_in[2];
    const float* Mud = (const float*)d_in[3];
    const float* Mdu = (const float*)d_in[4];
    const float* Auu = (const float*)d_in[5];
    const float* Add = (const float*)d_in[6];
    const float* Aud = (const float*)d_in[7];
    const float* Adu = (const float*)d_in[8];
    float* y = (float*)d_out;
    float* W = (float*)d_ws;   // needs 4*16*1024*192*4 = 48 MiB

    // 1) Qp precompute: 4*16*1024*96 threads
    k_qprime<<<24576, 256, 0, stream>>>(x, Muu, Mdd, Mud, Mdu, W);

    // 2) fused scores -> softmax -> AV (WMMA), one wave per (ntile, head, block)
    dim3 g2(64, 16, 4);
    k_attn<<<g2, 32, 0, stream>>>(x, W);

    // 3) combine with A / ksi, direct write of y (no atomics)
    k_combine<<<768, 256, 0, stream>>>(W, Auu, Add, Aud, Adu, y);
}